// Attention_72404558676364
// MI455X (gfx1250) — compile-verified
//
#include <hip/hip_runtime.h>

typedef __attribute__((ext_vector_type(2))) float v2f;
typedef __attribute__((ext_vector_type(8))) float v8f;

#define HDIM 1024
#define LDIM 512
#define BDIM 64
#define TWO_H 2048

// ---------------------------------------------------------------------------
// Kernel 1: w[j] = sum_k W_att[k][HDIM + j] * vec[k]   (j in [0, HDIM))
// Coalesced across j; vec[k] is wave-uniform (scalar-cached). 4 MB read.
// ---------------------------------------------------------------------------
__global__ void __launch_bounds__(256)
proj_w_kernel(const float* __restrict__ W_att,
              const float* __restrict__ vec,
              float* __restrict__ w)
{
    int j = blockIdx.x * blockDim.x + threadIdx.x;
    if (j >= HDIM) return;
    const float* col = W_att + HDIM + j;
    float acc = 0.f;
#pragma unroll 8
    for (int k = 0; k < HDIM; ++k) {
        acc = fmaf(col[(size_t)k * TWO_H], vec[k], acc);
    }
    w[j] = acc;
}

// ---------------------------------------------------------------------------
// Kernel 2: logits[i] = dot(hs[i, :], w)  for i = l*B + b  (32768 rows of 1024)
// One wave per 16-row tile, accumulated with V_WMMA_F32_16X16X4_F32.
//   A tile (16x4 f32, 2 VGPRs): lane (half=lane>>4, m=lane&15) holds
//       A[m, 2*half + {0,1}]  -> one float2 load per step.
//   B tile (4x16 f32, 2 VGPRs): B[k][n] = w[h0+k] for all n (broadcast cols);
//       same per-lane K mapping -> float2 load of w[h0 + 2*half + {0,1}].
//   All 16 D columns are identical = the 16 dot products; lanes 0 and 16
//   (D column 0) hold rows 0-7 / 8-15 in their 8 accumulator VGPRs.
// hs is a 128 MB single-pass stream: load it with TH=NT (nontemporal) so it
// doesn't displace anything in the 192 MB L2.
// ---------------------------------------------------------------------------
__global__ void __launch_bounds__(256)
attn_logits_kernel(const float* __restrict__ hs,
                   const float* __restrict__ w,
                   float* __restrict__ logits)
{
    __shared__ float w_lds[HDIM];
    const int tid = threadIdx.x;

    // Stage w (4 KB) into LDS, coalesced.
#pragma unroll
    for (int i = 0; i < HDIM; i += 256) w_lds[i + tid] = w[i + tid];
    __syncthreads();

    const int wave = tid >> 5;        // 0..7
    const int lane = tid & 31;
    const int half = lane >> 4;       // 0 or 1  -> K offset 0 or 2
    const int mrow = lane & 15;       // A-matrix row within tile

    const long tile    = (long)blockIdx.x * 8 + wave;
    const long rowBase = tile * 16;

    const float* aPtr = hs + (rowBase + mrow) * (long)HDIM + half * 2;
    const float* bPtr = &w_lds[half * 2];

    v8f acc = {};
#pragma unroll 8
    for (int k = 0; k < HDIM; k += 4) {
        v2f a = __builtin_nontemporal_load((const v2f*)(aPtr + k)); // streaming
        v2f b = *(const v2f*)(bPtr + k);   // w[k+2h : k+2h+2]  (ds_load_b64)
        acc = __builtin_amdgcn_wmma_f32_16x16x4_f32(
            /*neg_a=*/false, a, /*neg_b=*/false, b,
            /*c_mod=*/(short)0, acc, /*reuse_a=*/false, /*reuse_b=*/false);
    }

    // D column 0: lane 0 -> rows 0-7, lane 16 -> rows 8-15.
    if (mrow == 0) {
        float* dst = logits + rowBase + half * 8;
#pragma unroll
        for (int i = 0; i < 8; ++i) dst[i] = acc[i];
    }
}

// ---------------------------------------------------------------------------
// Kernel 3: out[b, 0, l] = softmax_l(logits[l*B + b]).  One block per b.
// ---------------------------------------------------------------------------
__global__ void __launch_bounds__(256)
softmax_kernel(const float* __restrict__ logits, float* __restrict__ out)
{
    __shared__ float red[256];
    const int b = blockIdx.x;
    const int tid = threadIdx.x;

    const float x0 = logits[(2 * tid + 0) * BDIM + b];
    const float x1 = logits[(2 * tid + 1) * BDIM + b];

    red[tid] = fmaxf(x0, x1);
    __syncthreads();
#pragma unroll
    for (int s = 128; s > 0; s >>= 1) {
        if (tid < s) red[tid] = fmaxf(red[tid], red[tid + s]);
        __syncthreads();
    }
    const float mx = red[0];
    __syncthreads();

    const float e0 = __expf(x0 - mx);
    const float e1 = __expf(x1 - mx);
    red[tid] = e0 + e1;
    __syncthreads();
#pragma unroll
    for (int s = 128; s > 0; s >>= 1) {
        if (tid < s) red[tid] += red[tid + s];
        __syncthreads();
    }
    const float inv = 1.0f / red[0];

    out[b * LDIM + 2 * tid + 0] = e0 * inv;
    out[b * LDIM + 2 * tid + 1] = e1 * inv;
}

// ---------------------------------------------------------------------------
// Host-side launcher. Inputs (setup_inputs order):
//   0: hidden (B,H)      -- unused: contributes only a per-b constant that
//   1: hs_encoder (L,B,H)   cancels under softmax over l (shift invariance)
//   2: W_att (H, 2H)
//   3: b_att (H)         -- unused: same cancellation
//   4: vector (H, 1)
// Output: (B, 1, L) float32.
// Workspace: w (HDIM floats) + logits (L*B floats) = 135 KB.
// ---------------------------------------------------------------------------
extern "C" void kernel_launch(void* const* d_in, const int* in_sizes, int n_in,
                              void* d_out, int out_size, void* d_ws, size_t ws_size,
                              hipStream_t stream)
{
    const float* hs    = (const float*)d_in[1];
    const float* W_att = (const float*)d_in[2];
    const float* vec   = (const float*)d_in[4];
    float* out = (float*)d_out;

    float* w      = (float*)d_ws;          // HDIM floats
    float* logits = w + HDIM;              // LDIM*BDIM floats

    proj_w_kernel<<<HDIM / 256, 256, 0, stream>>>(W_att, vec, w);
    attn_logits_kernel<<<(LDIM * BDIM) / (8 * 16), 256, 0, stream>>>(hs, w, logits);
    softmax_kernel<<<BDIM, 256, 0, stream>>>(logits, out);
}